// PredefinedPrototypesAngularLoss_29265907155138
// MI455X (gfx1250) — compile-verified
//
#include <hip/hip_runtime.h>

typedef float v2f __attribute__((ext_vector_type(2)));
typedef float v8f __attribute__((ext_vector_type(8)));

#define EPSF 1e-12f

// ---------------------------------------------------------------------------
// Kernel 1: recover the one-hot index per sample. One wave (32 lanes) scans
// one row of `labels` (n_classes floats); index reduced with shfl_xor max.
// ---------------------------------------------------------------------------
__global__ void find_idx_kernel(const float* __restrict__ labels,
                                int* __restrict__ idx, int n_classes) {
    const int wave = threadIdx.x >> 5;
    const int lane = threadIdx.x & 31;
    const int n = blockIdx.x * 8 + wave;
    const float* row = labels + (long long)n * n_classes;
    int best = 0;
    for (int c = lane; c < n_classes; c += 32)
        if (row[c] > 0.5f) best = c;
#pragma unroll
    for (int o = 16; o > 0; o >>= 1)
        best = max(best, __shfl_xor(best, o, 32));
    if (lane == 0) idx[n] = best;
}

// ---------------------------------------------------------------------------
// Kernel 2: per wave, a 16-sample tile. acc = sum_k A x B over embed, where
// A rows are feature rows and B columns are the gathered prototype rows, so
// acc[i][j] = F_i . P_j ; we keep the diagonal. Two v_wmma_f32_16x16x4_f32
// per 8 columns; float4 loads; norms accumulated in VALU from the same data.
// Column permutation (k + 4*hi) is identical for A and B => exact result.
// ---------------------------------------------------------------------------
__global__ void cos_tile_kernel(const float* __restrict__ F,
                                const float* __restrict__ P,
                                const int* __restrict__ idx,
                                float* __restrict__ partials, int embed) {
    __shared__ float ssum[8];
    const int wave = threadIdx.x >> 5;
    const int lane = threadIdx.x & 31;
    const int j    = lane & 15;   // sample within tile (A: M, B: N)
    const int hi   = lane >> 4;   // which K half this lane supplies

    const int s = (blockIdx.x * 8 + wave) * 16 + j;
    const int proto = idx[s];
    const float* frow = F + (long long)s * embed;
    const float* prow = P + (long long)proto * embed;

    v8f acc = {};
    float fn2 = 0.0f, pn2 = 0.0f;

    for (int k = 0; k < embed; k += 8) {
        const float4 f = *reinterpret_cast<const float4*>(frow + k + 4 * hi);
        const float4 p = *reinterpret_cast<const float4*>(prow + k + 4 * hi);

        v2f a0; a0.x = f.x; a0.y = f.y;
        v2f b0; b0.x = p.x; b0.y = p.y;
        acc = __builtin_amdgcn_wmma_f32_16x16x4_f32(false, a0, false, b0,
                                                    (short)0, acc, false, false);
        v2f a1; a1.x = f.z; a1.y = f.w;
        v2f b1; b1.x = p.z; b1.y = p.w;
        acc = __builtin_amdgcn_wmma_f32_16x16x4_f32(false, a1, false, b1,
                                                    (short)0, acc, false, false);

        fn2 = fmaf(f.x, f.x, fmaf(f.y, f.y, fmaf(f.z, f.z, fmaf(f.w, f.w, fn2))));
        pn2 = fmaf(p.x, p.x, fmaf(p.y, p.y, fmaf(p.z, p.z, fmaf(p.w, p.w, pn2))));
    }

    // lanes j and j+16 each hold half of sample j's squared norms
    fn2 += __shfl_xor(fn2, 16, 32);
    pn2 += __shfl_xor(pn2, 16, 32);

    // Diagonal of the 16x16 f32 C/D layout:
    //   sample j<8  -> lane j,      VGPR j
    //   sample j>=8 -> lane 16+j,   VGPR j-8      (i.e. lanes 24..31, VGPR lane&7)
    const int r = lane & 7;
    float dotv = 0.0f;
#pragma unroll
    for (int q = 0; q < 8; ++q)
        dotv = (q == r) ? acc[q] : dotv;

    const bool active = (lane < 8) || (lane >= 24);
    float nf = fmaxf(sqrtf(fn2), EPSF);
    float np = fmaxf(sqrtf(pn2), EPSF);
    float contrib = active ? (1.0f - dotv / (nf * np)) : 0.0f;

#pragma unroll
    for (int o = 16; o > 0; o >>= 1)
        contrib += __shfl_xor(contrib, o, 32);
    if (lane == 0) ssum[wave] = contrib;
    __syncthreads();
    if (threadIdx.x == 0) {
        float t = 0.0f;
        for (int w = 0; w < 8; ++w) t += ssum[w];
        partials[blockIdx.x] = t;
    }
}

// ---------------------------------------------------------------------------
// Kernel 3: deterministic final reduction of per-block partials -> mean.
// ---------------------------------------------------------------------------
__global__ void final_reduce_kernel(const float* __restrict__ partials,
                                    float* __restrict__ out, int nparts,
                                    float invN) {
    __shared__ float sm[256];
    sm[threadIdx.x] = (threadIdx.x < nparts) ? partials[threadIdx.x] : 0.0f;
    __syncthreads();
    for (int o = 128; o > 0; o >>= 1) {
        if ((int)threadIdx.x < o) sm[threadIdx.x] += sm[threadIdx.x + o];
        __syncthreads();
    }
    if (threadIdx.x == 0) out[0] = sm[0] * invN;
}

extern "C" void kernel_launch(void* const* d_in, const int* in_sizes, int n_in,
                              void* d_out, int out_size, void* d_ws, size_t ws_size,
                              hipStream_t stream) {
    const float* features   = (const float*)d_in[0];
    const float* labels     = (const float*)d_in[1];
    const float* mean_class = (const float*)d_in[2];

    const int embed     = 1024;
    const int N         = in_sizes[0] / embed;   // 32768
    const int n_classes = in_sizes[1] / N;       // 1000

    int*   idx      = (int*)d_ws;
    float* partials = (float*)((char*)d_ws + (size_t)N * sizeof(int));

    // 1) one-hot index recovery: one wave per sample
    find_idx_kernel<<<N / 8, 256, 0, stream>>>(labels, idx, n_classes);

    // 2) WMMA tile kernel: 128 samples per block (8 waves x 16 samples)
    const int nblocks = N / 128;                 // 256
    cos_tile_kernel<<<nblocks, 256, 0, stream>>>(features, mean_class, idx,
                                                 partials, embed);

    // 3) deterministic mean
    final_reduce_kernel<<<1, 256, 0, stream>>>(partials, (float*)d_out,
                                               nblocks, 1.0f / (float)N);
}